// MaxsatLayer_12979391169080
// MI455X (gfx1250) — compile-verified
//
#include <hip/hip_runtime.h>
#include <math.h>

#define N_VARS 64
#define NV1    65          // N_VARS + 1
#define M_CL   256
#define KD     12
#define GPAD   80          // padded Gram dim = 5*16
#define PI_F   3.14159265358979323846f

#define WAVES_PER_BLOCK 4
#define BLOCK2 (WAVES_PER_BLOCK * 32)

typedef float v2f __attribute__((ext_vector_type(2)));
typedef float v8f __attribute__((ext_vector_type(8)));

// ---------------------------------------------------------------------------
// Kernel 1: Gram matrix G = Sw^T * Sw  (Sw = S * w[:,None]), padded to 80x80.
// One wave per 16x16 output tile; K=256 reduced in chunks of 4 with
// V_WMMA_F32_16X16X4_F32.
// A (16x4 f32) lane layout (ISA 7.12.2): lanes 0-15 hold M=0..15,
//   VGPR0 = K0 (lanes 0-15) / K2 (lanes 16-31), VGPR1 = K1 / K3.
// B (4x16) mirrors this with N striped across lanes.
// ---------------------------------------------------------------------------
__global__ __launch_bounds__(32) void gram_wmma_kernel(
    const float* __restrict__ S, const float* __restrict__ w,
    float* __restrict__ Gpad) {
  const int lane = threadIdx.x;       // 0..31
  const int ti   = blockIdx.x / 5;    // output row tile
  const int tj   = blockIdx.x % 5;    // output col tile
  const int half = lane >> 4;
  const int r16  = lane & 15;
  const int n1   = ti * 16 + r16;     // A row index (Gram row)
  const int n2   = tj * 16 + r16;     // B col index (Gram col)

  v8f c = {};
  for (int m0 = 0; m0 < M_CL; m0 += 4) {
    const int ka = m0 + 2 * half;     // this lane covers K rows {ka, ka+1}
    const float w0 = w[ka];
    const float w1 = w[ka + 1];
    float a0 = 0.0f, a1 = 0.0f, b0 = 0.0f, b1 = 0.0f;
    if (n1 < NV1) {
      a0 = S[ka * NV1 + n1] * w0;
      a1 = S[(ka + 1) * NV1 + n1] * w1;
    }
    if (n2 < NV1) {
      b0 = S[ka * NV1 + n2] * w0;
      b1 = S[(ka + 1) * NV1 + n2] * w1;
    }
    v2f a; a.x = a0; a.y = a1;
    v2f b; b.x = b0; b.y = b1;
    c = __builtin_amdgcn_wmma_f32_16x16x4_f32(
        /*neg_a=*/false, a, /*neg_b=*/false, b,
        /*c_mod=*/(short)0, c, /*reuse_a=*/false, /*reuse_b=*/false);
  }

#pragma unroll
  for (int v = 0; v < 8; ++v) {
    const int rowM = v + 8 * half;    // C/D layout: VGPR v -> M = v (+8 for hi lanes)
    Gpad[(ti * 16 + rowM) * GPAD + tj * 16 + r16] = c[v];
  }
}

// ---------------------------------------------------------------------------
// Kernel 2: per-batch Gauss-Seidel sweeps on the Gram matrix.
// One wave32 per batch element; G shared in LDS per block; V per-wave in LDS.
// ---------------------------------------------------------------------------
__global__ __launch_bounds__(BLOCK2) void mixsat_main_kernel(
    const float* __restrict__ z, const float* __restrict__ r,
    const float* __restrict__ Gpad, const int* __restrict__ max_iter_p,
    float* __restrict__ out) {
  __shared__ float Gs[NV1 * NV1];                 // 16.9 KB
  __shared__ float Vs[WAVES_PER_BLOCK][NV1 * KD]; // 12.5 KB

  const int tid = threadIdx.x;

  // Stage G into LDS (block-cooperative)
  for (int idx = tid; idx < NV1 * NV1; idx += BLOCK2) {
    const int gi = idx / NV1;
    const int gj = idx - gi * NV1;
    Gs[idx] = Gpad[gi * GPAD + gj];
  }

  const int wave = tid >> 5;
  const int lane = tid & 31;
  const int b    = blockIdx.x * WAVES_PER_BLOCK + wave;
  float* V = Vs[wave];

  // ---- init V ----
  // Row 0 = e0
  if (lane < KD) V[lane] = (lane == 0) ? 1.0f : 0.0f;
  // Rows 1..64: V[i] = -cos(pi z) e0 + sin(pi z) * r_perp_normalized
#pragma unroll
  for (int rr = 0; rr < 2; ++rr) {
    const int i  = 1 + lane + 32 * rr;            // 1..64
    const int n  = i - 1;
    const float ang = PI_F * z[b * N_VARS + n];
    const float sn  = sinf(ang);
    const float cs  = cosf(ang);
    float rp[KD];
    rp[0] = 0.0f;
    float ss = 0.0f;
#pragma unroll
    for (int kk = 1; kk < KD; ++kk) {
      const float rv = r[(b * N_VARS + n) * KD + kk];
      rp[kk] = rv;
      ss += rv * rv;
    }
    const float inv = 1.0f / fmaxf(sqrtf(ss), 1e-8f);
    V[i * KD + 0] = -cs;
#pragma unroll
    for (int kk = 1; kk < KD; ++kk) V[i * KD + kk] = sn * rp[kk] * inv;
  }
  __syncthreads();

  const int iters = *max_iter_p;

  const int k   = lane % KD;     // vector component this lane owns
  const int h   = lane / KD;     // j-parity half (0/1 active; 2 = idle lanes)
  const bool act = (lane < 2 * KD);

  for (int it = 0; it < iters; ++it) {
    for (int i = 1; i < NV1; ++i) {
      const float* Grow = &Gs[i * NV1];
      // partial g[k] over j = h, h+2, ... (includes j==i; diag removed below)
      float g = 0.0f;
      if (act) {
        for (int j = h; j < NV1; j += 2) {
          g = fmaf(Grow[j], V[j * KD + k], g);
        }
      }
      // combine the two j-halves into lanes 0..11
      const int src = (lane + KD) & 31;
      const float gc = g + __shfl(g, src, 32);
      const float vold = V[i * KD + k];
      const float gii  = Grow[i];
      const float gf   = gc - gii * vold;          // remove j==i term
      // squared norm across the 12 components (xor tree over the wave)
      float sq = (lane < KD) ? gf * gf : 0.0f;
      sq += __shfl_xor(sq, 16, 32);
      sq += __shfl_xor(sq, 8, 32);
      sq += __shfl_xor(sq, 4, 32);
      sq += __shfl_xor(sq, 2, 32);
      sq += __shfl_xor(sq, 1, 32);
      const float denom = fmaxf(sqrtf(sq), 1e-8f);
      const float vnew  = -gf / denom;
      if (lane < KD) V[i * KD + lane] = vnew;
      // wave-private LDS RAW across lanes: force store completion + compiler order
      asm volatile("s_wait_dscnt 0" ::: "memory");
    }
  }

  // ---- output: arccos(clip(-V[1:,0])) / pi ----
#pragma unroll
  for (int rr = 0; rr < 2; ++rr) {
    const int i = 1 + lane + 32 * rr;
    float c0 = -V[i * KD + 0];
    c0 = fminf(fmaxf(c0, -1.0f + 1e-6f), 1.0f - 1e-6f);
    out[b * N_VARS + (i - 1)] = acosf(c0) / PI_F;
  }
}

// ---------------------------------------------------------------------------
extern "C" void kernel_launch(void* const* d_in, const int* in_sizes, int n_in,
                              void* d_out, int out_size, void* d_ws, size_t ws_size,
                              hipStream_t stream) {
  const float* z        = (const float*)d_in[0];   // (B, 64)
  const float* S        = (const float*)d_in[1];   // (256, 65)
  const float* w        = (const float*)d_in[2];   // (256,)
  const float* r        = (const float*)d_in[3];   // (B, 64, 12)
  const int*   max_iter = (const int*)d_in[4];     // scalar

  float* Gpad = (float*)d_ws;                      // 80*80 f32 = 25.6 KB
  const int B = in_sizes[0] / N_VARS;

  // 1) Gram matrix via WMMA f32 16x16x4 (25 tiles, one wave each)
  gram_wmma_kernel<<<25, 32, 0, stream>>>(S, w, Gpad);

  // 2) Batched Gauss-Seidel sweeps, one wave per batch element
  mixsat_main_kernel<<<B / WAVES_PER_BLOCK, BLOCK2, 0, stream>>>(
      z, r, Gpad, max_iter, (float*)d_out);
}